// GeometryEngine_23063974379791
// MI455X (gfx1250) — compile-verified
//
#include <hip/hip_runtime.h>

typedef float v2f __attribute__((ext_vector_type(2)));
typedef float v8f __attribute__((ext_vector_type(8)));

#define SPB     128          // sites per block (4 waves)
#define GSTRIDE 66           // floats per site in LDS Gamma (64 + pad, 264B = 8B aligned)
#define DSTRIDE 17           // padded row stride for the 16x16 D tile
#define DXF     0.01f
#define REGF    1e-4f

// symmetrized metric load: g[i][j] = 0.5*(m[n,i,j] + m[n,j,i])
__device__ __forceinline__ void sym_load(const float* __restrict__ mf, int n, float g[4][4]) {
    const float* p = mf + (size_t)n * 16;
    #pragma unroll
    for (int i = 0; i < 4; ++i)
        #pragma unroll
        for (int j = 0; j < 4; ++j)
            g[i][j] = 0.5f * (p[i * 4 + j] + p[j * 4 + i]);
}

// Compute Gamma^l_{mv} for site n, write 64 floats [l*16+m*4+v] into LDS dst.
__device__ void gamma_site(const float* __restrict__ mf, int n, int N, float* __restrict__ dst) {
    // finite difference of symmetrized metric along the lattice (fd1 semantics)
    const int  ln  = (n == 0)     ? 0     : n - 1;
    const int  rn  = (n == N - 1) ? N - 1 : n + 1;
    const float fac = (n == 0 || n == N - 1) ? (1.0f / DXF) : (0.5f / DXF);

    float gl[4][4], gr[4][4], dg[4][4], m[4][4], gi[4][4], H[4][4];
    sym_load(mf, ln, gl);
    sym_load(mf, rn, gr);
    #pragma unroll
    for (int i = 0; i < 4; ++i)
        #pragma unroll
        for (int j = 0; j < 4; ++j)
            dg[i][j] = (gr[i][j] - gl[i][j]) * fac;

    sym_load(mf, n, m);
    #pragma unroll
    for (int i = 0; i < 4; ++i) m[i][i] += REGF;

    // 4x4 inverse via adjugate / sub-determinants
    float A2323 = m[2][2]*m[3][3] - m[2][3]*m[3][2];
    float A1323 = m[2][1]*m[3][3] - m[2][3]*m[3][1];
    float A1223 = m[2][1]*m[3][2] - m[2][2]*m[3][1];
    float A0323 = m[2][0]*m[3][3] - m[2][3]*m[3][0];
    float A0223 = m[2][0]*m[3][2] - m[2][2]*m[3][0];
    float A0123 = m[2][0]*m[3][1] - m[2][1]*m[3][0];
    float A2313 = m[1][2]*m[3][3] - m[1][3]*m[3][2];
    float A1313 = m[1][1]*m[3][3] - m[1][3]*m[3][1];
    float A1213 = m[1][1]*m[3][2] - m[1][2]*m[3][1];
    float A2312 = m[1][2]*m[2][3] - m[1][3]*m[2][2];
    float A1312 = m[1][1]*m[2][3] - m[1][3]*m[2][1];
    float A1212 = m[1][1]*m[2][2] - m[1][2]*m[2][1];
    float A0313 = m[1][0]*m[3][3] - m[1][3]*m[3][0];
    float A0213 = m[1][0]*m[3][2] - m[1][2]*m[3][0];
    float A0312 = m[1][0]*m[2][3] - m[1][3]*m[2][0];
    float A0212 = m[1][0]*m[2][2] - m[1][2]*m[2][0];
    float A0113 = m[1][0]*m[3][1] - m[1][1]*m[3][0];
    float A0112 = m[1][0]*m[2][1] - m[1][1]*m[2][0];

    float det = m[0][0]*(m[1][1]*A2323 - m[1][2]*A1323 + m[1][3]*A1223)
              - m[0][1]*(m[1][0]*A2323 - m[1][2]*A0323 + m[1][3]*A0223)
              + m[0][2]*(m[1][0]*A1323 - m[1][1]*A0323 + m[1][3]*A0123)
              - m[0][3]*(m[1][0]*A1223 - m[1][1]*A0223 + m[1][2]*A0123);
    float rd = 1.0f / det;

    gi[0][0] = rd *  (m[1][1]*A2323 - m[1][2]*A1323 + m[1][3]*A1223);
    gi[0][1] = rd * -(m[0][1]*A2323 - m[0][2]*A1323 + m[0][3]*A1223);
    gi[0][2] = rd *  (m[0][1]*A2313 - m[0][2]*A1313 + m[0][3]*A1213);
    gi[0][3] = rd * -(m[0][1]*A2312 - m[0][2]*A1312 + m[0][3]*A1212);
    gi[1][0] = rd * -(m[1][0]*A2323 - m[1][2]*A0323 + m[1][3]*A0223);
    gi[1][1] = rd *  (m[0][0]*A2323 - m[0][2]*A0323 + m[0][3]*A0223);
    gi[1][2] = rd * -(m[0][0]*A2313 - m[0][2]*A0313 + m[0][3]*A0213);
    gi[1][3] = rd *  (m[0][0]*A2312 - m[0][2]*A0312 + m[0][3]*A0212);
    gi[2][0] = rd *  (m[1][0]*A1323 - m[1][1]*A0323 + m[1][3]*A0123);
    gi[2][1] = rd * -(m[0][0]*A1323 - m[0][1]*A0323 + m[0][3]*A0123);
    gi[2][2] = rd *  (m[0][0]*A1313 - m[0][1]*A0313 + m[0][3]*A0113);
    gi[2][3] = rd * -(m[0][0]*A1312 - m[0][1]*A0312 + m[0][3]*A0112);
    gi[3][0] = rd * -(m[1][0]*A1223 - m[1][1]*A0223 + m[1][2]*A0123);
    gi[3][1] = rd *  (m[0][0]*A1223 - m[0][1]*A0223 + m[0][2]*A0123);
    gi[3][2] = rd * -(m[0][0]*A1213 - m[0][1]*A0213 + m[0][2]*A0113);
    gi[3][3] = rd *  (m[0][0]*A1212 - m[0][1]*A0212 + m[0][2]*A0112);

    // H = g^{-1} @ dg
    #pragma unroll
    for (int l = 0; l < 4; ++l)
        #pragma unroll
        for (int v = 0; v < 4; ++v) {
            float acc = 0.0f;
            #pragma unroll
            for (int s = 0; s < 4; ++s) acc += gi[l][s] * dg[s][v];
            H[l][v] = acc;
        }

    // Gamma^l_{mv} = 0.5*(d_{m==1} H[l][v] + d_{v==1} H[l][m] - gi[l][1]*dg[m][v])
    #pragma unroll
    for (int l = 0; l < 4; ++l)
        #pragma unroll
        for (int mm = 0; mm < 4; ++mm)
            #pragma unroll
            for (int v = 0; v < 4; ++v) {
                float t = -gi[l][1] * dg[mm][v];
                if (mm == 1) t += H[l][v];
                if (v  == 1) t += H[l][mm];
                dst[l * 16 + mm * 4 + v] = 0.5f * t;
            }
}

__global__ __launch_bounds__(SPB) void ricci_kernel(const float* __restrict__ mf,
                                                    float* __restrict__ out, int N) {
    __shared__ float sGam[(SPB + 2) * GSTRIDE];   // Gamma for sites [base-1 .. base+SPB]
    __shared__ float sD[(SPB / 32) * 16 * DSTRIDE]; // per-wave 16x16 D tile

    const int t    = threadIdx.x;
    const int base = blockIdx.x * SPB;

    // ---- Phase 1: Christoffels into LDS (with halo) ----
    {
        int n = base + t; if (n > N - 1) n = N - 1;
        gamma_site(mf, n, N, &sGam[(t + 1) * GSTRIDE]);
        if (t == 0) {
            int nh = base - 1; if (nh < 0) nh = 0;       // unused when base==0
            gamma_site(mf, nh, N, &sGam[0]);
        }
        if (t == SPB - 1) {
            int nh = base + SPB; if (nh > N - 1) nh = N - 1; // unused at right edge
            gamma_site(mf, nh, N, &sGam[(SPB + 1) * GSTRIDE]);
        }
    }
    __syncthreads();

    // ---- Phase 2: one v_wmma_f32_16x16x4_f32 per site (wave-cooperative) ----
    const int wave = t >> 5;
    const int lane = t & 31;
    float* Dw = &sD[wave * 16 * DSTRIDE];
    const int c    = lane & 15;      // column index = 4*v + s
    const int vI   = c >> 2;
    const int sI   = c & 3;
    const int aoff = (lane & 15) * 4 + (lane >> 4) * 2;  // A: row=(r,m)=lane%16, K pair by half
    const int boff = (lane >> 4) * 32 + (lane & 15);     // B: rows l0,l0+1 (l0=0 or 2), col=lane%16
    const int rowb = (lane >> 4) * 8;                    // D: rows p / p+8 per half

    #pragma unroll 1
    for (int i = 0; i < 32; ++i) {
        const int ts = wave * 32 + i;            // site within block
        const float* G = &sGam[(ts + 1) * GSTRIDE];

        v2f a, b;
        a[0] = G[aoff];     a[1] = G[aoff + 1];
        b[0] = G[boff];     b[1] = G[boff + 16];
        v8f d = {0.f, 0.f, 0.f, 0.f, 0.f, 0.f, 0.f, 0.f};
        // P[(r,m),(v,s)] = sum_l Gamma[r][m][l] * Gamma[l][v][s]
        d = __builtin_amdgcn_wmma_f32_16x16x4_f32(false, a, false, b,
                                                  (short)0, d, false, false);
        __builtin_amdgcn_wave_barrier();
        #pragma unroll
        for (int p = 0; p < 8; ++p) Dw[(p + rowb) * DSTRIDE + c] = d[p];
        __builtin_amdgcn_wave_barrier();

        // quad terms: +sum_r P[(r,r),(v,s)]  (rows 0,5,10,15),  -sum_r P[(r,v),(r,s)]
        float rq = Dw[0 * DSTRIDE + c] + Dw[5 * DSTRIDE + c]
                 + Dw[10 * DSTRIDE + c] + Dw[15 * DSTRIDE + c];
        #pragma unroll
        for (int r = 0; r < 4; ++r) rq -= Dw[(4 * r + vI) * DSTRIDE + (4 * r + sI)];

        // derivative terms: d1 Gamma^1_{vs} - delta_{v,1} * sum_r d1 Gamma^r_{rs}
        const int n2 = base + ts;
        const int li = ts + 1;
        const int lL = li - 1 + ((n2 == 0) ? 1 : 0);
        const int lR = li + 1 - ((n2 == N - 1) ? 1 : 0);
        const float fac = (n2 == 0 || n2 == N - 1) ? (1.0f / DXF) : (0.5f / DXF);
        const float* GL = &sGam[lL * GSTRIDE];
        const float* GR = &sGam[lR * GSTRIDE];

        float ric = fac * (GR[16 + 4 * vI + sI] - GL[16 + 4 * vI + sI]);
        if (vI == 1) {
            #pragma unroll
            for (int r = 0; r < 4; ++r)
                ric -= fac * (GR[20 * r + sI] - GL[20 * r + sI]);
        }
        ric += rq;

        if (lane < 16 && n2 < N)
            out[(size_t)n2 * 16 + sI * 4 + vI] = ric;
    }
}

extern "C" void kernel_launch(void* const* d_in, const int* in_sizes, int n_in,
                              void* d_out, int out_size, void* d_ws, size_t ws_size,
                              hipStream_t stream) {
    (void)n_in; (void)out_size; (void)d_ws; (void)ws_size;
    const float* mf = (const float*)d_in[0];
    float* out = (float*)d_out;
    const int N = in_sizes[0] / 16;           // [N,4,4]
    const int blocks = (N + SPB - 1) / SPB;   // N=131072 -> 1024 full blocks
    ricci_kernel<<<blocks, SPB, 0, stream>>>(mf, out, N);
}